// GraphSAGEXBatNorm_22428319220707
// MI455X (gfx1250) — compile-verified
//
#include <hip/hip_runtime.h>

#define NFEAT 128
#define NHID  64
#define NCLASS 40
#define BN_EPS 1e-5f

typedef __attribute__((ext_vector_type(2))) float v2f;
typedef __attribute__((ext_vector_type(8))) float v8f;

// ---------------------------------------------------------------------------
// Degree: deg[dst] += 1  (float atomics; exact for deg < 2^24)
// ---------------------------------------------------------------------------
__global__ void deg_kernel(const int* __restrict__ dst, float* __restrict__ deg, int nE) {
    int e = blockIdx.x * blockDim.x + threadIdx.x;
    if (e < nE) atomicAdd(&deg[dst[e]], 1.0f);
}

// ---------------------------------------------------------------------------
// Scatter-add aggregation: agg[dst, :] += X[src, :]
// One thread per (edge, feature); feature-major for coalescing.
// L2-resident working set -> global_atomic_add_f32 is the right path.
// ---------------------------------------------------------------------------
template <int D>
__global__ void scatter_add_kernel(const float* __restrict__ X,
                                   const int* __restrict__ src,
                                   const int* __restrict__ dst,
                                   float* __restrict__ agg, int nE) {
    long long t = (long long)blockIdx.x * blockDim.x + threadIdx.x;
    if (t >= (long long)nE * D) return;
    int e = (int)(t / D);
    int f = (int)(t % D);
    int s = src[e];
    int d = dst[e];
    atomicAdd(agg + (size_t)d * D + f, X[(size_t)s * D + f]);
}

// ---------------------------------------------------------------------------
// Fused SAGE layer: Out = epilogue( (Aagg/deg) @ Wl + Aself @ Wr + bias )
// One wave32 per 16-node tile; f32 WMMA 16x16x4, K-loop over KDIM.
//
// A layout (32-bit A 16x4): lane r=lane&15 is row M=r; lanes 0-15 hold K
// pair {k,k+1}, lanes 16-31 hold {k+2,k+3} (VGPR j = K offset j).
// B layout mirrored: col = lane&15, same K split.
// C/D: VGPR j -> row j (lanes 0-15) / row j+8 (lanes 16-31), col = lane&15.
//
// Columns >= NOUT (only layer 3: cols 40..47) accumulate garbage in their
// own D columns (WMMA columns are independent) and are simply never stored;
// B loads are address-clamped to stay in-bounds.
// ---------------------------------------------------------------------------
template <int KDIM, int NCB, int NOUT, bool BN, bool RELU>
__global__ __launch_bounds__(32)
void sage_gemm_kernel(const float* __restrict__ Aagg,
                      const float* __restrict__ Aself,
                      const float* __restrict__ deg,
                      const float* __restrict__ Wl,
                      const float* __restrict__ Wr,
                      const float* __restrict__ bias,
                      const float* __restrict__ bn_g,
                      const float* __restrict__ bn_b,
                      const float* __restrict__ bn_m,
                      const float* __restrict__ bn_v,
                      float* __restrict__ Out,
                      int nrows) {
    const int tile  = blockIdx.x;
    const int lane  = threadIdx.x;          // 0..31
    const int r     = lane & 15;            // row within tile (A), col within block (B)
    const int khalf = (lane >> 4) << 1;     // 0 or 2: K sub-offset for this half-wave
    const int row   = tile * 16 + r;
    const int rowc  = (row < nrows) ? row : (nrows - 1);

    const float dg   = deg[rowc];
    const float invd = 1.0f / fmaxf(dg, 1.0f);

    const float* __restrict__ arow_agg  = Aagg  + (size_t)rowc * KDIM;
    const float* __restrict__ arow_self = Aself + (size_t)rowc * KDIM;

    // Per-col-block B base pointers (column clamp folded in, hoisted out of K loop)
    const float* wlp[NCB];
    const float* wrp[NCB];
#pragma unroll
    for (int cb = 0; cb < NCB; ++cb) {
        const int col  = cb * 16 + r;
        const int colc = (col < NOUT) ? col : (NOUT - 1);
        wlp[cb] = Wl + colc;
        wrp[cb] = Wr + colc;
    }

    v8f acc[NCB];
#pragma unroll
    for (int cb = 0; cb < NCB; ++cb)
#pragma unroll
        for (int j = 0; j < 8; ++j) acc[cb][j] = 0.0f;

    for (int k = 0; k < KDIM; k += 4) {
        const int kk = k + khalf;           // even => 8B aligned
        v2f aA = *(const v2f*)(arow_agg + kk);
        aA.x *= invd; aA.y *= invd;
        const v2f aS = *(const v2f*)(arow_self + kk);

#pragma unroll
        for (int cb = 0; cb < NCB; ++cb) {
            v2f bL, bR;
            bL.x = wlp[cb][(size_t)kk * NOUT];
            bL.y = wlp[cb][(size_t)(kk + 1) * NOUT];
            bR.x = wrp[cb][(size_t)kk * NOUT];
            bR.y = wrp[cb][(size_t)(kk + 1) * NOUT];
            acc[cb] = __builtin_amdgcn_wmma_f32_16x16x4_f32(
                false, aA, false, bL, (short)0, acc[cb], false, false);
            acc[cb] = __builtin_amdgcn_wmma_f32_16x16x4_f32(
                false, aS, false, bR, (short)0, acc[cb], false, false);
        }
    }

    // Epilogue + store. D layout: row = tile*16 + j + (lane>=16 ? 8 : 0).
    const int  rowbase  = tile * 16 + (khalf << 2);        // +0 or +8
    const bool fullTile = (tile * 16 + 16) <= nrows;       // wave-uniform
#pragma unroll
    for (int cb = 0; cb < NCB; ++cb) {
        const int col = cb * 16 + r;
        if (col >= NOUT) continue;                         // only trims layer-3 cb=2 lanes
        const float bval = bias[col];
        float scale = 1.0f, mean = 0.0f, shift = 0.0f;
        if (BN) {
            scale = bn_g[col] * rsqrtf(bn_v[col] + BN_EPS);
            mean  = bn_m[col];
            shift = bn_b[col];
        }
        float* __restrict__ outp = Out + (size_t)rowbase * NOUT + col;
        if (fullTile) {
#pragma unroll
            for (int j = 0; j < 8; ++j) {
                float val = acc[cb][j] + bval;
                if (BN)   val = (val - mean) * scale + shift;
                if (RELU) val = fmaxf(val, 0.0f);
                outp[(size_t)j * NOUT] = val;
            }
        } else {
#pragma unroll
            for (int j = 0; j < 8; ++j) {
                if (rowbase + j >= nrows) continue;
                float val = acc[cb][j] + bval;
                if (BN)   val = (val - mean) * scale + shift;
                if (RELU) val = fmaxf(val, 0.0f);
                outp[(size_t)j * NOUT] = val;
            }
        }
    }
}

// ---------------------------------------------------------------------------
// Host-side orchestration (all on `stream`, graph-capture safe)
// ---------------------------------------------------------------------------
extern "C" void kernel_launch(void* const* d_in, const int* in_sizes, int n_in,
                              void* d_out, int out_size, void* d_ws, size_t ws_size,
                              hipStream_t stream) {
    const float* x    = (const float*)d_in[0];
    const int*   ei   = (const int*)  d_in[1];
    const float* w1l  = (const float*)d_in[2];
    const float* w1r  = (const float*)d_in[3];
    const float* b1   = (const float*)d_in[4];
    const float* wxl  = (const float*)d_in[5];
    const float* wxr  = (const float*)d_in[6];
    const float* bx   = (const float*)d_in[7];
    const float* w2l  = (const float*)d_in[8];
    const float* w2r  = (const float*)d_in[9];
    const float* b2   = (const float*)d_in[10];
    const float* bn3g = (const float*)d_in[11];
    const float* bn3b = (const float*)d_in[12];
    const float* bn3m = (const float*)d_in[13];
    const float* bn3v = (const float*)d_in[14];
    const float* bn2g = (const float*)d_in[15];
    const float* bn2b = (const float*)d_in[16];
    const float* bn2m = (const float*)d_in[17];
    const float* bn2v = (const float*)d_in[18];

    const int nN = in_sizes[0] / NFEAT;   // 100000
    const int nE = in_sizes[1] / 2;       // 1600000
    const int* src = ei;
    const int* dst = ei + nE;

    // Workspace layout (~103 MB)
    float* ws  = (float*)d_ws;
    float* deg = ws;                                  // [nN]
    float* agg = deg + nN;                            // [nN * NFEAT] (reused; layers 2/3 use first nN*NHID)
    float* h1  = agg + (size_t)nN * NFEAT;            // [nN * NHID]
    float* h2  = h1  + (size_t)nN * NHID;             // [nN * NHID]

    const int ntiles = (nN + 15) / 16;

    // --- degrees (shared by all layers) ---
    hipMemsetAsync(deg, 0, (size_t)nN * sizeof(float), stream);
    deg_kernel<<<(nE + 255) / 256, 256, 0, stream>>>(dst, deg, nE);

    // --- layer 1: h1 = relu(mean_agg(x) @ w1l + x @ w1r + b1) ---
    hipMemsetAsync(agg, 0, (size_t)nN * NFEAT * sizeof(float), stream);
    {
        long long tot = (long long)nE * NFEAT;
        scatter_add_kernel<NFEAT><<<(int)((tot + 255) / 256), 256, 0, stream>>>(x, src, dst, agg, nE);
    }
    sage_gemm_kernel<NFEAT, 4, NHID, false, true><<<ntiles, 32, 0, stream>>>(
        agg, x, deg, w1l, w1r, b1, nullptr, nullptr, nullptr, nullptr, h1, nN);

    // --- layer 2: h2 = relu(bn3(mean_agg(h1) @ wxl + h1 @ wxr + bx)) ---
    hipMemsetAsync(agg, 0, (size_t)nN * NHID * sizeof(float), stream);
    {
        long long tot = (long long)nE * NHID;
        scatter_add_kernel<NHID><<<(int)((tot + 255) / 256), 256, 0, stream>>>(h1, src, dst, agg, nE);
    }
    sage_gemm_kernel<NHID, 4, NHID, true, true><<<ntiles, 32, 0, stream>>>(
        agg, h1, deg, wxl, wxr, bx, bn3g, bn3b, bn3m, bn3v, h2, nN);

    // --- layer 3: out = bn2(mean_agg(h2) @ w2l + h2 @ w2r + b2) ---
    hipMemsetAsync(agg, 0, (size_t)nN * NHID * sizeof(float), stream);
    {
        long long tot = (long long)nE * NHID;
        scatter_add_kernel<NHID><<<(int)((tot + 255) / 256), 256, 0, stream>>>(h2, src, dst, agg, nE);
    }
    sage_gemm_kernel<NHID, 3, NCLASS, true, false><<<ntiles, 32, 0, stream>>>(
        agg, h2, deg, w2l, w2r, b2, bn2g, bn2b, bn2m, bn2v, (float*)d_out, nN);
}